// IndexedLinear_85779086836294
// MI455X (gfx1250) — compile-verified
//
#include <hip/hip_runtime.h>

typedef __attribute__((ext_vector_type(2))) float v2f;
typedef __attribute__((ext_vector_type(4))) float v4f;
typedef __attribute__((ext_vector_type(8))) float v8f;

#define S_  8
#define U_  32
#define V_  32
#define C_  32
#define Z_  65536
#define INW  (S_ * U_)   // 256 floats per input2 row
#define OUTW (S_ * V_)   // 256 floats per output row
#define LDS_STRIDE 36    // 16B-aligned rows, conflict-free column reads

__global__ __launch_bounds__(256)
void indexed_linear_wmma(const float* __restrict__ Wt,     // (C, S*U*V)
                         const float* __restrict__ X,      // (Z, S*U)
                         const int*   __restrict__ counts, // (C,)
                         const float* __restrict__ coef,   // (S,)
                         float* __restrict__ out)          // (Z, S*V)
{
    __shared__ __align__(16) float xs[S_][16 * LDS_STRIDE];

    const int lane = threadIdx.x & 31;
    const int s    = threadIdx.x >> 5;      // one wave per s
    const int z0   = blockIdx.x * 16;       // 16-row tile
    const int h    = lane >> 4;             // lane half (0/1)
    const int l16  = lane & 15;

    // ---- Stage X tile (16 rows x 32 cols for this s) into LDS, coalesced ----
    float* xw = &xs[s][0];
#pragma unroll
    for (int i = 0; i < 4; ++i) {
        int e   = i * 128 + lane * 4;       // flat element in 16x32 slab
        int row = e >> 5;
        int col = e & 31;
        v4f v = *(const v4f*)(X + (size_t)(z0 + row) * INW + s * U_ + col);
        *(v4f*)(xw + row * LDS_STRIDE + col) = v;
    }

    // ---- Per-row segment id from counts (rows are sorted by segment) ----
    const int myrow = z0 + l16;
    int seg = 0, acc = 0;
#pragma unroll
    for (int c = 0; c < C_; ++c) {
        acc += counts[c];
        seg += (myrow >= acc) ? 1 : 0;
    }

    v8f acc0 = {0.f,0.f,0.f,0.f,0.f,0.f,0.f,0.f};
    v8f acc1 = {0.f,0.f,0.f,0.f,0.f,0.f,0.f,0.f};

    // ---- Group loop over distinct segments inside the tile (uniform flow) ----
    unsigned remaining = 0xFFFFu;
    while (remaining) {
        int leader = __ffs(remaining) - 1;
        int g = __shfl(seg, leader, 32);
        unsigned long long bal = __ballot(seg == g);
        unsigned match = (unsigned)bal & 0xFFFFu;
        remaining &= ~match;

        // mask A rows not in this segment group (lane owns row l16)
        float amask = (seg == g) ? 1.0f : 0.0f;

        const float* Wg = Wt + (size_t)g * (S_ * U_ * V_) + (size_t)s * (U_ * V_);

#pragma unroll
        for (int kk = 0; kk < 8; ++kk) {
            int kbase = kk * 4 + 2 * h;     // this lane's K pair {kbase, kbase+1}

            // A fragment from LDS (consecutive cols, 8B aligned), masked
            v2f a = *(const v2f*)(xw + l16 * LDS_STRIDE + kbase);
            a *= amask;

            // B fragments from global (L2-resident weights)
            const float* wk = Wg + (size_t)kbase * V_ + l16;
            v2f b0; b0[0] = wk[0];  b0[1] = wk[V_];       // n-tile 0
            v2f b1; b1[0] = wk[16]; b1[1] = wk[V_ + 16];  // n-tile 1

            acc0 = __builtin_amdgcn_wmma_f32_16x16x4_f32(
                       false, a, false, b0, (short)0, acc0, false, false);
            acc1 = __builtin_amdgcn_wmma_f32_16x16x4_f32(
                       false, a, false, b1, (short)0, acc1, false, false);
        }
    }

    // ---- Scale by coefficient and store D (VGPR j -> row j + 8h, col l16) ----
    const float cf = coef[s];
    float* outp = out + (size_t)z0 * OUTW + s * V_;
#pragma unroll
    for (int j = 0; j < 8; ++j) {
        int m = j + 8 * h;
        outp[(size_t)m * OUTW + l16]      = cf * acc0[j];
        outp[(size_t)m * OUTW + 16 + l16] = cf * acc1[j];
    }
}

extern "C" void kernel_launch(void* const* d_in, const int* in_sizes, int n_in,
                              void* d_out, int out_size, void* d_ws, size_t ws_size,
                              hipStream_t stream) {
    const float* Wt     = (const float*)d_in[0];  // input1 (C, S*U*V)
    const float* X      = (const float*)d_in[1];  // input2 (Z, S*U)
    const int*   counts = (const int*)  d_in[2];  // (C,)
    const float* coef   = (const float*)d_in[3];  // (S,)
    float* out = (float*)d_out;                   // (Z, S*V)

    dim3 grid(Z_ / 16);   // 4096 tiles of 16 rows
    dim3 block(256);      // 8 waves, one per s
    hipLaunchKernelGGL(indexed_linear_wmma, grid, block, 0, stream,
                       Wt, X, counts, coef, out);
}